// RadialFreqGate_60988535603858
// MI455X (gfx1250) — compile-verified
//
#include <hip/hip_runtime.h>
#include <hip/hip_bf16.h>
#include <math.h>

typedef float v2f __attribute__((ext_vector_type(2)));
typedef float v4f __attribute__((ext_vector_type(4)));
typedef float v8f __attribute__((ext_vector_type(8)));

#define B_   4
#define C_   32
#define H_   512
#define W_   512
#define HT_  64
#define WT_  64
#define HW_  (H_ * W_)            // 262144
#define NPATCH_PER_B (HT_ * WT_)  // 4096
#define PI_F 3.14159265358979323846f

// ws layout (floats):
//   [0, 32768)          : A-fragment table, 8192 float4 entries (kstep,mt,lane)
//   [32768, 49152)      : edge_strength (B*4096)
//   [49152, 65536)      : edge_coeff    (B*4096)
#define WS_ATAB_F4   0
#define WS_ES        32768
#define WS_EC        (32768 + B_ * NPATCH_PER_B)

// ---------------------------------------------------------------------------
// Kernel 1: build the two 8x8 circular-convolution kernels K_lo/K_hi from the
// gain vector, then materialize the complete WMMA A-fragment table in the
// exact per-lane layout of V_WMMA_F32_16X16X4_F32:
//   entry[(kstep*4+mt)*32 + lane] = {Tlo[m][k], Tlo[m][k+1], Thi[m][k], Thi[m][k+1]}
// with m = mt*16 + (lane&15), k = kstep*4 + (lane>>4)*2, and
// T[m][k] = K(((m>>3)-(k>>3))&7, ((m&7)-(k&7))&7)   (circulant).
// This table is identical for every wave in the grid -> compute once.
// ---------------------------------------------------------------------------
__global__ __launch_bounds__(256) void RFG_buildK(const float* __restrict__ gain,
                                                  v4f* __restrict__ atab) {
    __shared__ float mlo[40];
    __shared__ float mhi[40];
    __shared__ float Klo[64];
    __shared__ float Khi[64];
    const int tid = threadIdx.x;
    if (tid < 40) {
        const int ky = tid / 5, kx = tid % 5;
        const float kyf = (float)(ky <= 4 ? ky : 8 - ky);
        const float rmax = sqrtf(32.0f) + 1e-8f;
        float r = sqrtf(kyf * kyf + (float)(kx * kx)) / rmax;
        r = fminf(fmaxf(r, 0.0f), 1.0f);
        float wv[6];
        float wsum = 0.0f;
        #pragma unroll
        for (int i = 0; i < 6; ++i) {
            const float c = (float)i * 0.2f;          // centers = i/5
            float w = 1.0f - fabsf(r - c) * 5.0f;     // 1 - |r-c|/DELTA
            w = fmaxf(w, 0.0f);
            wv[i] = w;
            wsum += w;
        }
        float m = 0.0f;
        #pragma unroll
        for (int i = 0; i < 6; ++i) m += (wv[i] / (wsum + 1e-8f)) * gain[i];
        m = fmaxf(m, 0.0f);
        const float hf = (r >= 0.6f) ? 1.0f : 0.0f;
        mlo[tid] = m * (1.0f - hf);
        mhi[tid] = m * hf;
    }
    __syncthreads();
    if (tid < 64) {
        const int dy = tid >> 3, dx = tid & 7;
        float alo = 0.0f, ahi = 0.0f;
        for (int ky = 0; ky < 8; ++ky) {
            const int kyf = (ky <= 4) ? ky : 8 - ky;
            for (int kx = 0; kx < 8; ++kx) {
                const int kxf = (kx <= 4) ? kx : 8 - kx;
                const float ang = (float)(ky * dy + kx * dx) * (PI_F * 0.25f);
                const float cv = cosf(ang);
                alo += mlo[kyf * 5 + kxf] * cv;
                ahi += mhi[kyf * 5 + kxf] * cv;
            }
        }
        Klo[tid] = alo * (1.0f / 64.0f);
        Khi[tid] = ahi * (1.0f / 64.0f);
    }
    __syncthreads();
    for (int i = tid; i < 8192; i += 256) {
        const int lane  = i & 31;
        const int mt    = (i >> 5) & 3;
        const int kstep = i >> 7;
        const int nIdx  = lane & 15;
        const int half  = lane >> 4;
        const int k  = kstep * 4 + half * 2;
        const int ky = k >> 3, kx = k & 7;
        const int m  = mt * 16 + nIdx;
        const int dy  = ((m >> 3) - ky) & 7;
        const int dx  = ((m & 7) - kx) & 7;
        const int dx2 = (dx - 1) & 7;
        const v4f e = {Klo[dy * 8 + dx], Klo[dy * 8 + dx2],
                       Khi[dy * 8 + dx], Khi[dy * 8 + dx2]};
        atab[i] = e;
    }
}

// ---------------------------------------------------------------------------
// Kernel 2: channel mean. xmean[b,h,w] = mean_c x[b,c,h,w]. Coalesced on w.
// ---------------------------------------------------------------------------
__global__ __launch_bounds__(256) void RFG_mean(const float* __restrict__ x,
                                                float* __restrict__ xmean) {
    const int idx = blockIdx.x * 256 + threadIdx.x;   // over B*H*W = 2^20
    const int b = idx >> 18;
    const int pix = idx & (HW_ - 1);
    const float* p = x + (size_t)b * C_ * HW_ + pix;
    float s = 0.0f;
    #pragma unroll
    for (int c = 0; c < C_; ++c) s += p[(size_t)c * HW_];
    xmean[idx] = s * (1.0f / 32.0f);
}

// ---------------------------------------------------------------------------
// Kernel 3: Sobel edge magnitude (zero-padded), averaged over each 8x8 patch.
// ---------------------------------------------------------------------------
__global__ __launch_bounds__(64) void RFG_edge(const float* __restrict__ xmean,
                                               float* __restrict__ es) {
    __shared__ float red[64];
    const int bi = blockIdx.x;               // B * 4096 blocks
    const int b = bi >> 12;
    const int p = bi & (NPATCH_PER_B - 1);
    const int ph = p >> 6, pw = p & 63;
    const int tid = threadIdx.x;
    const int gy = ph * 8 + (tid >> 3);
    const int gx = pw * 8 + (tid & 7);
    const float* mb = xmean + (size_t)b * HW_;
    float n[3][3];
    #pragma unroll
    for (int dy = -1; dy <= 1; ++dy)
        #pragma unroll
        for (int dx = -1; dx <= 1; ++dx) {
            const int yy = gy + dy, xx = gx + dx;
            n[dy + 1][dx + 1] = (yy >= 0 && yy < H_ && xx >= 0 && xx < W_)
                                    ? mb[yy * W_ + xx] : 0.0f;
        }
    const float gxv = (n[0][2] - n[0][0]) + 2.0f * (n[1][2] - n[1][0]) + (n[2][2] - n[2][0]);
    const float gyv = (n[2][0] - n[0][0]) + 2.0f * (n[2][1] - n[0][1]) + (n[2][2] - n[0][2]);
    red[tid] = sqrtf(gxv * gxv + gyv * gyv + 1e-6f);
    __syncthreads();
    #pragma unroll
    for (int s = 32; s > 0; s >>= 1) {
        if (tid < s) red[tid] += red[tid + s];
        __syncthreads();
    }
    if (tid == 0) es[bi] = red[0] * (1.0f / 64.0f);
}

// ---------------------------------------------------------------------------
// Kernel 4: per-batch min/max normalize -> edge_coeff = 1 + 0.5 * norm.
// ---------------------------------------------------------------------------
__global__ __launch_bounds__(256) void RFG_coeff(const float* __restrict__ es,
                                                 float* __restrict__ ec) {
    __shared__ float smn[256];
    __shared__ float smx[256];
    const int b = blockIdx.x, tid = threadIdx.x;
    const float* e = es + b * NPATCH_PER_B;
    float mn = 1e30f, mx = -1e30f;
    for (int i = tid; i < NPATCH_PER_B; i += 256) {
        const float v = e[i];
        mn = fminf(mn, v);
        mx = fmaxf(mx, v);
    }
    smn[tid] = mn; smx[tid] = mx;
    __syncthreads();
    #pragma unroll
    for (int s = 128; s > 0; s >>= 1) {
        if (tid < s) {
            smn[tid] = fminf(smn[tid], smn[tid + s]);
            smx[tid] = fmaxf(smx[tid], smx[tid + s]);
        }
        __syncthreads();
    }
    mn = smn[0]; mx = smx[0];
    const float inv = 0.5f / ((mx - mn) + 1e-6f);   // EDGE_BOOST / (range+eps)
    float* o = ec + b * NPATCH_PER_B;
    for (int i = tid; i < NPATCH_PER_B; i += 256) o[i] = 1.0f + (e[i] - mn) * inv;
}

// ---------------------------------------------------------------------------
// Kernel 5 (main): Out(64xN) = T_lo @ X + c * (T_hi @ X) via
// V_WMMA_F32_16X16X4_F32. Each wave: one 64(M) x 16(patch) tile,
// 16 ksteps x 4 mtiles x {lo,hi} = 128 wmma. A-fragments come from a
// precomputed LDS table (one ds_load_b128 per (kstep,mt) delivers both the
// lo and hi fragment, already in WMMA lane layout). Epilogue: each
// accumulator half is one contiguous patch row -> b128 stores.
// ---------------------------------------------------------------------------
__global__ __launch_bounds__(256) void RFG_gate(const float* __restrict__ x,
                                                const float* __restrict__ ws,
                                                float* __restrict__ out) {
    __shared__ v4f Atab[2048];               // 32 KB: [(kstep*4+mt)*32 + lane]
    {
        const v4f* atg = (const v4f*)ws;
        for (int i = threadIdx.x; i < 2048; i += 256) Atab[i] = atg[i];
    }
    __syncthreads();

    const float* coeff = ws + WS_EC;

    const int tid  = threadIdx.x;
    const int lane = tid & 31;
    const int wave = (blockIdx.x * 256 + tid) >> 5;      // 32768 wave-tiles
    const int n16  = wave & 3;                           // pw group (16 patches)
    const int ph   = (wave >> 2) & 63;
    const int ch   = (wave >> 8) & 31;
    const int b    = wave >> 13;

    const int nIdx = lane & 15;          // patch column within tile (N)
    const int half = lane >> 4;          // lane-half -> K split in A/B frags
    const int pw   = n16 * 16 + nIdx;

    const float cval = coeff[b * NPATCH_PER_B + ph * 64 + pw];

    const float* xbase = x   + ((size_t)(b * C_ + ch) * H_ + ph * 8) * W_;
    float*       obase = out + ((size_t)(b * C_ + ch) * H_ + ph * 8) * W_;

    v8f acc_lo[4], acc_hi[4];
    const v8f zero = {0.f, 0.f, 0.f, 0.f, 0.f, 0.f, 0.f, 0.f};
    #pragma unroll
    for (int mt = 0; mt < 4; ++mt) { acc_lo[mt] = zero; acc_hi[mt] = zero; }

    #pragma unroll
    for (int kstep = 0; kstep < 16; ++kstep) {
        const int k  = kstep * 4 + half * 2;   // this lane-half's K pair
        const int ky = k >> 3;                 // input row within patch
        const int kx = k & 7;                  // input col within patch
        // B fragment: X[k..k+1][patch nIdx] — 8-byte aligned pair load
        const v2f bfrag = *(const v2f*)(xbase + ky * W_ + pw * 8 + kx);
        #pragma unroll
        for (int mt = 0; mt < 4; ++mt) {
            // one b128 LDS load = both A fragments, already in lane layout
            const v4f a = Atab[(kstep * 4 + mt) * 32 + lane];
            acc_lo[mt] = __builtin_amdgcn_wmma_f32_16x16x4_f32(
                false, a.lo, false, bfrag, (short)0, acc_lo[mt], false, false);
            acc_hi[mt] = __builtin_amdgcn_wmma_f32_16x16x4_f32(
                false, a.hi, false, bfrag, (short)0, acc_hi[mt], false, false);
        }
    }

    // Epilogue: out = lo + c*hi. Accumulator j=0..7 of (mt, half) is exactly
    // output patch row (mt*2 + half), columns 0..7 -> two b128 stores per mt.
    #pragma unroll
    for (int mt = 0; mt < 4; ++mt) {
        v4f r0, r1;
        #pragma unroll
        for (int j = 0; j < 4; ++j) {
            r0[j] = acc_lo[mt][j]     + cval * acc_hi[mt][j];
            r1[j] = acc_lo[mt][j + 4] + cval * acc_hi[mt][j + 4];
        }
        float* rp = obase + (mt * 2 + half) * W_ + pw * 8;
        *(v4f*)rp       = r0;
        *((v4f*)rp + 1) = r1;
    }
}

// ---------------------------------------------------------------------------
extern "C" void kernel_launch(void* const* d_in, const int* in_sizes, int n_in,
                              void* d_out, int out_size, void* d_ws, size_t ws_size,
                              hipStream_t stream) {
    const float* x    = (const float*)d_in[0];   // (4,32,512,512) fp32
    const float* gain = (const float*)d_in[1];   // (1,6) fp32
    float* out = (float*)d_out;
    float* wsf = (float*)d_ws;

    v4f*   atab = (v4f*)(wsf + WS_ATAB_F4);
    float* es   = wsf + WS_ES;
    float* ec   = wsf + WS_EC;
    // xmean scratch lives in the first 4 MB of d_out; fully overwritten later.
    float* xmean = out;

    RFG_buildK<<<1, 256, 0, stream>>>(gain, atab);
    RFG_mean  <<<(B_ * HW_) / 256, 256, 0, stream>>>(x, xmean);
    RFG_edge  <<<B_ * NPATCH_PER_B, 64, 0, stream>>>(xmean, es);
    RFG_coeff <<<B_, 256, 0, stream>>>(es, ec);
    RFG_gate  <<<4096, 256, 0, stream>>>(x, wsf, out);
}